// Decoder_32727650795997
// MI455X (gfx1250) — compile-verified
//
#include <hip/hip_runtime.h>

typedef __attribute__((ext_vector_type(2))) float v2f;
typedef __attribute__((ext_vector_type(8))) float v8f;

#define D_FEAT 128
#define E_POS  320000

// One wave (32 lanes) computes 16 edge dot-products via V_WMMA_F32_16X16X4_F32.
//  A (16x4 f32): lane l holds row M=l%16, K = {0,1} (lanes 0-15) or {2,3} (lanes 16-31)
//  B (4x16 f32): lane l holds col N=l%16, K rows split the same way
//  -> per K-chunk of 4, each lane does ONE aligned global_load_b64 per operand.
// Accumulate 32 chunks (K=128), then read the diagonal of the 16x16 result.
__global__ __launch_bounds__(256) void edge_dot_wmma(
    const float* __restrict__ x,
    const int*   __restrict__ pos_idx,   // int64 buffer viewed as (lo,hi) dword pairs
    const int*   __restrict__ neg_idx,
    float*       __restrict__ out,
    int n_groups)
{
    const int lane = threadIdx.x & 31;
    const int wave = (blockIdx.x * blockDim.x + threadIdx.x) >> 5;
    if (wave >= n_groups) return;            // never taken with exact launch; keeps EXEC full

    const int e0 = wave * 16;                // first edge of this 16-edge group
    const int m  = lane & 15;                // which edge this lane serves
    const int e  = e0 + m;

    // Gather src/tar node ids (values < 2^31: read low dword of the int64)
    int srow, trow;
    if (e < E_POS) {                         // wave-uniform: E_POS is a multiple of 16
        srow = pos_idx[2 * e];
        trow = pos_idx[2 * (E_POS + e)];
    } else {
        const int en = e - E_POS;
        srow = neg_idx[2 * en];
        trow = neg_idx[2 * (E_POS + en)];
    }

    const int kq = (lane < 16) ? 0 : 2;      // K sub-offset for this half-wave
    const float* pa = x + (long)srow * D_FEAT + kq;   // A: rows of src
    const float* pb = x + (long)trow * D_FEAT + kq;   // B: rows of tar (= columns of B)

    v8f acc = {};
#pragma unroll
    for (int c = 0; c < D_FEAT / 4; ++c) {
        v2f a = *(const v2f*)(pa + 4 * c);   // global_load_b64 (8B aligned)
        v2f b = *(const v2f*)(pb + 4 * c);   // global_load_b64 (8B aligned)
        // 8 args: (neg_a, A, neg_b, B, c_mod, C, reuse_a, reuse_b)
        acc = __builtin_amdgcn_wmma_f32_16x16x4_f32(
            false, a, false, b, (short)0, acc, false, false);
    }

    // Diagonal extract: D[i][i] -> VGPR (i&7) of lane i (i<8) / lane i+8 (i>=8).
    const int idx = lane & 7;
    float d = acc[0];
    d = (idx == 1) ? acc[1] : d;
    d = (idx == 2) ? acc[2] : d;
    d = (idx == 3) ? acc[3] : d;
    d = (idx == 4) ? acc[4] : d;
    d = (idx == 5) ? acc[5] : d;
    d = (idx == 6) ? acc[6] : d;
    d = (idx == 7) ? acc[7] : d;

    const bool active = (lane < 8) || (lane >= 16 && lane < 24);
    const int  row_m  = (lane < 16) ? lane : (lane - 8);
    if (active) out[e0 + row_m] = d;         // [E,1] output, contiguous floats
}

extern "C" void kernel_launch(void* const* d_in, const int* in_sizes, int n_in,
                              void* d_out, int out_size, void* d_ws, size_t ws_size,
                              hipStream_t stream) {
    const float* x   = (const float*)d_in[0];
    const int*   pos = (const int*)d_in[1];   // int64 data, read as dword pairs
    const int*   neg = (const int*)d_in[2];
    float*       out = (float*)d_out;

    const int E        = out_size;            // 640000
    const int n_groups = E / 16;              // 40000 (E is a multiple of 16)
    const int waves_per_block = 256 / 32;     // wave32
    const int blocks = (n_groups + waves_per_block - 1) / waves_per_block;

    edge_dot_wmma<<<blocks, 256, 0, stream>>>(x, pos, neg, out, n_groups);
}